// MLMMoE_15058155340405
// MI455X (gfx1250) — compile-verified
//
#include <hip/hip_runtime.h>

typedef __attribute__((ext_vector_type(2))) float v2f;
typedef __attribute__((ext_vector_type(4))) float v4f;
typedef __attribute__((ext_vector_type(8))) float v8f;
typedef __attribute__((ext_vector_type(4))) unsigned int v4u;
typedef __attribute__((ext_vector_type(8))) int v8i;
typedef __attribute__((ext_vector_type(4))) int v4i;

#ifndef __has_builtin
#define __has_builtin(x) 0
#endif
#if __has_builtin(__builtin_amdgcn_tensor_load_to_lds) && \
    __has_builtin(__builtin_amdgcn_s_wait_tensorcnt)
#define USE_TDM 1
#else
#define USE_TDM 0
#endif

#define TOK     16
#define DIN     256
#define NE      8
#define NT      4
#define H0      64
#define H1      32
#define GH      32

// padded LDS row strides (floats): +4 so row-stride % 64 banks != 0
#define SX_LD   260   // 256+4  (matches TDM pad: 256 DW interval, 4 DW pad)
#define SE0_LD  516   // 512+4
#define SG0H_LD 260
#define SG1H_LD 132   // 128+4
#define SX1_LD  516
#define SE1_LD  260

__device__ __forceinline__ v4f s4(float v) { v4f r = {v, v, v, v}; return r; }

// One 16(M tokens) x 16(N cols) tile, K-deep accumulation with
// V_WMMA_F32_16X16X4_F32. A from padded LDS, B streams from global
// (L2-resident weights). Fused bias + ReLU, result -> LDS.
__device__ __forceinline__ void wmma_tile(
    const float* __restrict__ sA, int aLD, int acol0, int kdim,
    const float* __restrict__ W, int ldw,          // W pre-offset to n0 column
    const float* __restrict__ bias,                // pre-offset to n0
    float* __restrict__ sOut, int oLD, int ocol0,
    int lane)
{
  const int n    = lane & 15;        // N column within tile
  const int koff = (lane >> 4) * 2;  // K sub-offset per f32 WMMA layout

  v8f acc = {};
  const float* arow = sA + (lane & 15) * aLD + acol0 + koff;
  const float* wp   = W + (size_t)koff * ldw + n;

  #pragma unroll 8
  for (int k = 0; k < kdim; k += 4) {
    v2f a, b;
    a.x = arow[k];                     // K = k+koff   (VGPR0)
    a.y = arow[k + 1];                 // K = k+koff+1 (VGPR1)
    const float* bp = wp + (size_t)k * ldw;
    b.x = bp[0];                       // B[K=k+koff][n]
    b.y = bp[ldw];                     // B[K=k+koff+1][n]
    acc = __builtin_amdgcn_wmma_f32_16x16x4_f32(
        false, a, false, b, (short)0, acc, false, false);
  }

  const float bv = bias[n];
  const int mh = (lane >> 4) * 8;
  #pragma unroll
  for (int r = 0; r < 8; ++r) {
    float v = acc[r] + bv;
    v = v > 0.0f ? v : 0.0f;           // ReLU
    sOut[(size_t)(mh + r) * oLD + ocol0 + n] = v;
  }
}

__global__ void __launch_bounds__(128) mlmmoe_kernel(
    const float* __restrict__ x,
    const float* __restrict__ We0,  const float* __restrict__ be0,
    const float* __restrict__ We1,  const float* __restrict__ be1,
    const float* __restrict__ Wg0a, const float* __restrict__ bg0a,
    const float* __restrict__ Wg0b, const float* __restrict__ bg0b,
    const float* __restrict__ Wg1a, const float* __restrict__ bg1a,
    const float* __restrict__ Wg1b, const float* __restrict__ bg1b,
    float* __restrict__ out)
{
  __shared__ float sX  [TOK * SX_LD];
  __shared__ float sE0 [TOK * SE0_LD];
  __shared__ float sG0H[TOK * SG0H_LD];
  __shared__ float sG1H[TOK * SG1H_LD];
  __shared__ float sX1 [TOK * SX1_LD];
  __shared__ float sE1 [TOK * SE1_LD];
  __shared__ float sWg0b[NE * GH * NE];   // 2048
  __shared__ float sbg0b[NE * NE];        // 64
  __shared__ float sWg1b[NT * GH * NE];   // 1024
  __shared__ float sbg1b[NT * NE];        // 32

  const int tid  = threadIdx.x;
  const int lane = tid & 31;
  const int wave = tid >> 5;
  const int b0   = blockIdx.x * TOK;

#if USE_TDM
  // --- x tile via Tensor Data Mover: 2D tile 256x16 fp32, HW-padded rows ---
  if (wave == 0) {
    const unsigned long long ga =
        (unsigned long long)(uintptr_t)(x + (size_t)b0 * DIN);
    const unsigned int ldsOff =
        (unsigned int)(uintptr_t)(const void*)sX;   // flat low32 == LDS offset

    v4u g0;
    g0.x = 1u;                                      // count=1 valid descriptor
    g0.y = ldsOff;                                  // lds_addr (bytes)
    g0.z = (unsigned int)ga;                        // global_addr[31:0]
    g0.w = ((unsigned int)(ga >> 32) & 0x01FFFFFFu) // global_addr[56:32]
         | (2u << 30);                              // type=2 ("image")
    v8i g1;
    g1[0] = (int)((2u << 16)      // data_size = 4B
                | (1u << 20)      // pad_enable
                | (7u << 22)      // pad_interval: 256 DWORDs (one row)
                | (3u << 25));    // pad_amount:   4 DWORDs  -> stride 260
    g1[1] = (int)(256u << 16);    // tensor_dim0 = 256 (bits 79:48, low half)
    g1[2] = 0;                    // dim0 hi / tensor_dim1 low16 (65536 -> 0)
    g1[3] = (int)(1u | (256u << 16)); // tensor_dim1 hi=1 | tile_dim0 = 256
    g1[4] = 16;                   // tile_dim1 = 16, tile_dim2 = 0
    g1[5] = 256;                  // tensor_dim0_stride = 256 (low32)
    g1[6] = 0;
    g1[7] = 0;
    v4i gz4 = {0, 0, 0, 0};       // groups 2/3 unused (2D tensor)
    v8i gz8 = {0, 0, 0, 0, 0, 0, 0, 0};
    __builtin_amdgcn_tensor_load_to_lds(g0, g1, gz4, gz4, gz8, 0);
    __builtin_amdgcn_s_wait_tensorcnt((short)0);    // TENSORcnt is per-wave
  }
#else
  // --- fallback: cooperative float4 staging ---
  for (int c = tid; c < TOK * (DIN / 4); c += 128) {
    const int r = c >> 6;
    const int q = c & 63;
    v4f v = *(const v4f*)(x + (size_t)(b0 + r) * DIN + q * 4);
    *(v4f*)(sX + (size_t)r * SX_LD + q * 4) = v;
  }
#endif

  // --- stage small gate weights (hot across all 16 tokens) ---
  for (int i = tid; i < NE * GH * NE; i += 128) sWg0b[i] = Wg0b[i];
  for (int i = tid; i < NE * NE;      i += 128) sbg0b[i] = bg0b[i];
  for (int i = tid; i < NT * GH * NE; i += 128) sWg1b[i] = Wg1b[i];
  for (int i = tid; i < NT * NE;      i += 128) sbg1b[i] = bg1b[i];
  __syncthreads();

  // ============ Phase A: level-0 GEMM on WMMA (56 N-tiles of 16) ============
  // cols: e0 (8*64=512) | g0h (8*32=256) | g1h (4*32=128)
  for (int t = wave; t < 56; t += 4) {
    const float* W; int ldw; const float* bias; float* op; int oLD; int cb;
    if (t < 32) {                      // e0 expert tiles
      const int e = t >> 2, n0 = (t & 3) * 16;
      W = We0 + (size_t)e * DIN * H0 + n0; ldw = H0;
      bias = be0 + e * H0 + n0;
      op = sE0; oLD = SE0_LD; cb = e * H0 + n0;
    } else if (t < 48) {               // g0 hidden tiles
      const int g = (t - 32) >> 1, n0 = (t & 1) * 16;
      W = Wg0a + (size_t)g * DIN * GH + n0; ldw = GH;
      bias = bg0a + g * GH + n0;
      op = sG0H; oLD = SG0H_LD; cb = g * GH + n0;
    } else {                           // g1 hidden tiles
      const int tt = (t - 48) >> 1, n0 = (t & 1) * 16;
      W = Wg1a + (size_t)tt * DIN * GH + n0; ldw = GH;
      bias = bg1a + tt * GH + n0;
      op = sG1H; oLD = SG1H_LD; cb = tt * GH + n0;
    }
    __builtin_prefetch(W, 0, 1);       // global_prefetch_b8 (L2-hot weights)
    wmma_tile(sX, SX_LD, 0, DIN, W, ldw, bias, op, oLD, cb, lane);
  }
  __syncthreads();

  // ============ Phase B1: g0 softmax + x1 = g0 . e0 (vectorized VALU) ======
  {
    const int tok = tid >> 3;          // 16 tokens
    const int g   = tid & 7;           // 8 gates
    const float* wg = sWg0b + (size_t)g * GH * NE;          // [h][e], ld=8
    const float* gh = sG0H + (size_t)tok * SG0H_LD + g * GH;
    v4f l0 = *(const v4f*)(sbg0b + g * NE);
    v4f l1 = *(const v4f*)(sbg0b + g * NE + 4);
    #pragma unroll
    for (int h = 0; h < GH; h += 4) {
      v4f av = *(const v4f*)(gh + h);
      l0 += s4(av.x) * (*(const v4f*)(wg + (h + 0) * NE));
      l1 += s4(av.x) * (*(const v4f*)(wg + (h + 0) * NE + 4));
      l0 += s4(av.y) * (*(const v4f*)(wg + (h + 1) * NE));
      l1 += s4(av.y) * (*(const v4f*)(wg + (h + 1) * NE + 4));
      l0 += s4(av.z) * (*(const v4f*)(wg + (h + 2) * NE));
      l1 += s4(av.z) * (*(const v4f*)(wg + (h + 2) * NE + 4));
      l0 += s4(av.w) * (*(const v4f*)(wg + (h + 3) * NE));
      l1 += s4(av.w) * (*(const v4f*)(wg + (h + 3) * NE + 4));
    }
    float l[NE] = {l0.x, l0.y, l0.z, l0.w, l1.x, l1.y, l1.z, l1.w};
    float m = l[0];
    #pragma unroll
    for (int e = 1; e < NE; ++e) m = fmaxf(m, l[e]);
    float s = 0.0f;
    #pragma unroll
    for (int e = 0; e < NE; ++e) { l[e] = __expf(l[e] - m); s += l[e]; }
    const float inv = 1.0f / s;
    #pragma unroll
    for (int e = 0; e < NE; ++e) l[e] *= inv;

    const float* e0p = sE0 + (size_t)tok * SE0_LD;
    float*       x1p = sX1 + (size_t)tok * SX1_LD + g * H0;
    #pragma unroll 4
    for (int h = 0; h < H0; h += 4) {
      v4f acc = {0.0f, 0.0f, 0.0f, 0.0f};
      #pragma unroll
      for (int e = 0; e < NE; ++e)
        acc += s4(l[e]) * (*(const v4f*)(e0p + e * H0 + h));
      *(v4f*)(x1p + h) = acc;
    }
  }
  __syncthreads();

  // ============ Phase A2: e1 = relu(x1 . We1 + be1) on WMMA ============
  // 8 experts x 2 N-tiles, K = 64
  for (int t = wave; t < 16; t += 4) {
    const int e = t >> 1, n0 = (t & 1) * 16;
    const float* W    = We1 + (size_t)e * H0 * H1 + n0;     // [h][k], ld=32
    const float* bias = be1 + e * H1 + n0;
    wmma_tile(sX1, SX1_LD, e * H0, H0, W, H1, bias,
              sE1, SE1_LD, e * H1 + n0, lane);
  }
  __syncthreads();

  // ============ Phase B2: g1 softmax + out = g1 . e1 (vectorized) ==========
  {
    const int tok  = tid >> 3;
    const int j    = tid & 7;
    const int tt   = j >> 1;           // task 0..3
    const int half = j & 1;            // which 16 of H1=32
    const float* wt = sWg1b + (size_t)tt * GH * NE;
    const float* gh = sG1H + (size_t)tok * SG1H_LD + tt * GH;
    v4f l0 = *(const v4f*)(sbg1b + tt * NE);
    v4f l1 = *(const v4f*)(sbg1b + tt * NE + 4);
    #pragma unroll
    for (int h = 0; h < GH; h += 4) {
      v4f av = *(const v4f*)(gh + h);
      l0 += s4(av.x) * (*(const v4f*)(wt + (h + 0) * NE));
      l1 += s4(av.x) * (*(const v4f*)(wt + (h + 0) * NE + 4));
      l0 += s4(av.y) * (*(const v4f*)(wt + (h + 1) * NE));
      l1 += s4(av.y) * (*(const v4f*)(wt + (h + 1) * NE + 4));
      l0 += s4(av.z) * (*(const v4f*)(wt + (h + 2) * NE));
      l1 += s4(av.z) * (*(const v4f*)(wt + (h + 2) * NE + 4));
      l0 += s4(av.w) * (*(const v4f*)(wt + (h + 3) * NE));
      l1 += s4(av.w) * (*(const v4f*)(wt + (h + 3) * NE + 4));
    }
    float l[NE] = {l0.x, l0.y, l0.z, l0.w, l1.x, l1.y, l1.z, l1.w};
    float m = l[0];
    #pragma unroll
    for (int e = 1; e < NE; ++e) m = fmaxf(m, l[e]);
    float s = 0.0f;
    #pragma unroll
    for (int e = 0; e < NE; ++e) { l[e] = __expf(l[e] - m); s += l[e]; }
    const float inv = 1.0f / s;
    #pragma unroll
    for (int e = 0; e < NE; ++e) l[e] *= inv;

    const float* e1p = sE1 + (size_t)tok * SE1_LD + half * 16;
    float* op = out + (size_t)(b0 + tok) * (NT * H1) + tt * H1 + half * 16;
    #pragma unroll
    for (int q = 0; q < 4; ++q) {
      v4f acc = {0.0f, 0.0f, 0.0f, 0.0f};
      #pragma unroll
      for (int e = 0; e < NE; ++e)
        acc += s4(l[e]) * (*(const v4f*)(e1p + e * H1 + q * 4));
      *(v4f*)(op + q * 4) = acc;
    }
  }
}

extern "C" void kernel_launch(void* const* d_in, const int* in_sizes, int n_in,
                              void* d_out, int out_size, void* d_ws, size_t ws_size,
                              hipStream_t stream) {
  const float* x    = (const float*)d_in[0];
  const float* We0  = (const float*)d_in[1];
  const float* be0  = (const float*)d_in[2];
  const float* We1  = (const float*)d_in[3];
  const float* be1  = (const float*)d_in[4];
  const float* Wg0a = (const float*)d_in[5];
  const float* bg0a = (const float*)d_in[6];
  const float* Wg0b = (const float*)d_in[7];
  const float* bg0b = (const float*)d_in[8];
  const float* Wg1a = (const float*)d_in[9];
  const float* bg1a = (const float*)d_in[10];
  const float* Wg1b = (const float*)d_in[11];
  const float* bg1b = (const float*)d_in[12];
  float* outp = (float*)d_out;

  const int B = in_sizes[0] / DIN;     // 65536
  dim3 grid(B / TOK), block(128);
  hipLaunchKernelGGL(mlmmoe_kernel, grid, block, 0, stream,
                     x, We0, be0, We1, be1, Wg0a, bg0a, Wg0b, bg0b,
                     Wg1a, bg1a, Wg1b, bg1b, outp);
  (void)d_ws; (void)ws_size; (void)out_size; (void)n_in;
}